// SlotDiversityLoss_22703197126896
// MI455X (gfx1250) — compile-verified
//
#include <hip/hip_runtime.h>
#include <hip/hip_bf16.h>

typedef float v2f __attribute__((ext_vector_type(2)));
typedef float v8f __attribute__((ext_vector_type(8)));

// eps = 1e-8 on the norm  <=>  eps^2 = 1e-16 on the squared norm (normal f32)
#define COSEPS_SQ 1e-16f

// ---------------------------------------------------------------------------
// Zero the 512-float scratch (pair_sum[256] | pair_cnt[256]).
// ---------------------------------------------------------------------------
__global__ void sdl_zero_ws(float* __restrict__ ws) {
  int i = threadIdx.x;
  if (i < 512) ws[i] = 0.0f;
}

// ---------------------------------------------------------------------------
// Main streaming kernel. One wave32 per 16-slot timestep tile per iteration:
//   - 16 vec3 (one global_load_b96/lane) + 16 mask bytes loaded
//   - v_rsq normalize; masked-out rows zeroed IN THE OPERAND, so the Gram
//     matrix is pre-masked and pair_sum needs only max(0)+add per element
//   - WMMA #1: sims = U_masked * U_masked^T            (C = 0)
//   - WMMA #2: cnt += m * m^T, accumulated in the WMMA C operand across
//     the whole tile loop (rank-1 outer product, K0 = 0/1 mask)
// Flush once per wave via a clause of global f32 atomics.
// ---------------------------------------------------------------------------
__global__ __launch_bounds__(256) void sdl_main(
    const float* __restrict__ doa,          // [nTiles, 16, 3]
    const unsigned char* __restrict__ mask, // [nTiles, 16] (jax bool = 1B)
    float* __restrict__ ws,                 // [512] sums | counts
    int nTiles)
{
  const int lane  = threadIdx.x & 31;
  const int wavesPerBlock = blockDim.x >> 5;
  const int wave  = blockIdx.x * wavesPerBlock + (threadIdx.x >> 5);
  const int nWaves = gridDim.x * wavesPerBlock;
  const int row   = lane & 15;          // matrix row (A) / column (B,C)
  const bool hi   = lane >= 16;

  // contiguous chunk per wave: sequential HBM stream, prefetch-friendly
  const int per = (nTiles + nWaves - 1) / nWaves;
  const int t0  = wave * per;
  const int t1  = (t0 + per < nTiles) ? (t0 + per) : nTiles;

  float sum[8];
#pragma unroll
  for (int r = 0; r < 8; ++r) sum[r] = 0.0f;
  v8f cntacc = {};                      // pair_cnt accumulated by WMMA #2

  for (int tile = t0; tile < t1; ++tile) {
    // ---- load this lane's row vector (both halves load the same 16 rows;
    //      the 192B tile spans 1-2 cache lines, WGP$ absorbs duplication) ---
    const float* vp = doa + (size_t)tile * 48 + row * 3;
    float x = vp[0], y = vp[1], z = vp[2];
    unsigned char m = mask[(size_t)tile * 16 + row];

    // Unconditional, branchless prefetch of the next tile in this wave's
    // sequential stream; address clamped to the chunk's last tile so it is
    // always in-bounds (no divergent EXEC manipulation around the WMMAs).
    {
      int pfTile = tile + 1;
      pfTile = (pfTile < t1 - 1) ? pfTile : (t1 - 1);
      __builtin_prefetch(doa + (size_t)pfTile * 48 + row * 3, 0, 3);
    }

    // ---- normalize: u = v * rsqrt(max(||v||^2, eps^2))  (v_rsq_f32, TRANS)
    //      and zero the row entirely if mask is off -> Gram pre-masked ------
    float n2    = x * x + y * y + z * z;
    float inv   = __builtin_amdgcn_rsqf(fmaxf(n2, COSEPS_SQ));
    float scale = m ? inv : 0.0f;

    // ---- symmetric operand: lo lanes {K0=x,K1=y}, hi lanes {K2=z,K3=0} ----
    v2f ab;
    ab.x = (hi ? z : x) * scale;
    ab.y = hi ? 0.0f : (y * scale);

    // ---- 0/1 mask vector, K0 only: serves as both A and B of WMMA #2 -----
    v2f mv;
    mv.x = (!hi && m) ? 1.0f : 0.0f;
    mv.y = 0.0f;

    // ---- WMMA #1: sims = U*U^T (pre-masked), fresh C = 0 ------------------
    v8f c = {};
    c = __builtin_amdgcn_wmma_f32_16x16x4_f32(
        false, ab, false, ab, (short)0, c, false, false);

    // ---- WMMA #2: cntacc += m*m^T (accumulator chained across tiles) ------
    cntacc = __builtin_amdgcn_wmma_f32_16x16x4_f32(
        false, mv, false, mv, (short)0, cntacc, false, false);

    // ---- pair_sum: clamp-at-0 and accumulate; no mask logic needed --------
#pragma unroll
    for (int r = 0; r < 8; ++r) {
      sum[r] += fmaxf(c[r], 0.0f);
    }
  }

  // ---- flush per-wave accumulators: pair (M = r + 8*hi, N = row) ----
#pragma unroll
  for (int r = 0; r < 8; ++r) {
    const int M   = r + (hi ? 8 : 0);
    const int idx = M * 16 + row;
    atomicAdd(&ws[idx], sum[r]);         // global_atomic_add_f32
    atomicAdd(&ws[256 + idx], cntacc[r]);
  }
}

// ---------------------------------------------------------------------------
// Final reduction: mean over upper-triangle pairs that have any count.
// ---------------------------------------------------------------------------
__global__ __launch_bounds__(256) void sdl_final(
    const float* __restrict__ ws, float* __restrict__ out)
{
  __shared__ float sTot[256];
  __shared__ float sVal[256];
  const int t = threadIdx.x;        // pair index: i = t/16, j = t%16
  const int i = t >> 4;
  const int j = t & 15;

  float mean = 0.0f, valid = 0.0f;
  if (i < j) {                      // strict upper triangle
    const float c = ws[256 + t];
    if (c > 0.0f) {
      mean  = ws[t] / fmaxf(c, 1.0f);
      valid = 1.0f;
    }
  }
  sTot[t] = mean;
  sVal[t] = valid;
  __syncthreads();
  for (int s = 128; s > 0; s >>= 1) {
    if (t < s) { sTot[t] += sTot[t + s]; sVal[t] += sVal[t + s]; }
    __syncthreads();
  }
  if (t == 0) out[0] = sTot[0] / fmaxf(sVal[0], 1.0f);
}

// ---------------------------------------------------------------------------
extern "C" void kernel_launch(void* const* d_in, const int* in_sizes, int n_in,
                              void* d_out, int out_size, void* d_ws, size_t ws_size,
                              hipStream_t stream) {
  const float*         doa  = (const float*)d_in[0];          // [B,T,16,3] f32
  const unsigned char* mask = (const unsigned char*)d_in[1];  // [B,T,16] bool
  float* ws  = (float*)d_ws;
  float* out = (float*)d_out;

  const int nTiles = in_sizes[0] / 48;   // B*T timestep tiles of 16 vec3s

  sdl_zero_ws<<<1, 512, 0, stream>>>(ws);

  // 2048 blocks x 8 waves = 16384 waves -> ~16 contiguous tiles/wave at
  // B*T = 262144. Purely bandwidth-bound (~52 MiB stream @ 23.3 TB/s).
  const int blocks = 2048;
  sdl_main<<<blocks, 256, 0, stream>>>(doa, mask, ws, nTiles);

  sdl_final<<<1, 256, 0, stream>>>(ws, out);
}